// MultiBoxLoss_69733089018172
// MI455X (gfx1250) — compile-verified
//
#include <hip/hip_runtime.h>
#include <stdint.h>

// ---------------- problem constants (match reference) ----------------
#define THRESH 0.5f
#define NEG_POS 3
#define VAR0 0.1f
#define VAR1 0.2f

constexpr int B   = 16;
constexpr int P   = 100000;
constexpr int T   = 32;
constexpr int NID = 7;
constexpr int TPB = 256;                    // 8 wave32 waves per block
constexpr int GX  = (P + TPB - 1) / TPB;    // 391 blocks along P
constexpr int NWAVE = TPB / 32;

// ---------------- CDNA5 async global->LDS copy (ASYNCcnt path) ----------------
__device__ __forceinline__ void async_copy_b128(uint32_t lds_off, uint64_t gaddr) {
  // VDST = LDS byte address (workgroup-relative, low 32 bits of generic ptr),
  // VADDR = 64-bit global address, saddr = off  (GV mode)
  asm volatile("global_load_async_to_lds_b128 %0, %1, off"
               :: "v"(lds_off), "v"(gaddr) : "memory");
}
__device__ __forceinline__ void async_wait0() {
  asm volatile("s_wait_asynccnt 0" ::: "memory");
}

__device__ __forceinline__ float block_reduce_sum(float v, float* red) {
  red[threadIdx.x] = v;
  __syncthreads();
  for (int s = TPB / 2; s > 0; s >>= 1) {
    if ((int)threadIdx.x < s) red[threadIdx.x] += red[threadIdx.x + s];
    __syncthreads();
  }
  float r = red[0];
  __syncthreads();
  return r;
}

// ---------------- kernel 0: zero integer accumulators ----------------
__global__ void k_zero(int* __restrict__ num_pos) {
  if (threadIdx.x < B) num_pos[threadIdx.x] = 0;
}

// ---------------- kernel 1: fused matching ----------------
// One pass over priors per batch. Produces:
//   ovl/bti  : per-prior max IoU / argmax truth (first occurrence)
//   pval/pidx: per-(block,truth) best prior candidate (for global argmax)
__global__ void k_match(const float* __restrict__ targets,
                        const float* __restrict__ priors,
                        float* __restrict__ ovl,
                        int* __restrict__ bti,
                        float* __restrict__ pval,
                        int* __restrict__ pidx) {
  const int b     = blockIdx.y;
  const int chunk = blockIdx.x;
  const int p     = chunk * TPB + threadIdx.x;
  const int lane  = threadIdx.x & 31;
  const int wave  = threadIdx.x >> 5;

  __shared__ __align__(16) float st[T * 7];     // 224 floats = 56 x b128
  __shared__ float wv[T][NWAVE];
  __shared__ int   wi[T][NWAVE];

  if (threadIdx.x < (T * 7 * 4) / 16) {
    uint32_t lds = (uint32_t)(uintptr_t)(&st[0]) + threadIdx.x * 16u;
    uint64_t g   = (uint64_t)(uintptr_t)(targets + (size_t)b * T * 7) + threadIdx.x * 16u;
    async_copy_b128(lds, g);
  }
  async_wait0();
  __syncthreads();

  const bool valid = (p < P);
  float px0 = 0.f, py0 = 0.f, px1 = 0.f, py1 = 0.f, areaB = 0.f;
  if (valid) {
    float cx = priors[4 * p + 0], cy = priors[4 * p + 1];
    float w  = priors[4 * p + 2], h  = priors[4 * p + 3];
    px0 = cx - 0.5f * w; py0 = cy - 0.5f * h;
    px1 = cx + 0.5f * w; py1 = cy + 0.5f * h;
    areaB = w * h;
  }

  float bestv = -1.0f; int besti = 0;            // max over truths (this prior)
  for (int t = 0; t < T; ++t) {
    float iou = -1.0f;
    if (valid) {
      const float* tr = &st[t * 7];
      float ax0 = tr[2], ay0 = tr[3], ax1 = tr[4], ay1 = tr[5];
      float iw = fmaxf(fminf(ax1, px1) - fmaxf(ax0, px0), 0.0f);
      float ih = fmaxf(fminf(ay1, py1) - fmaxf(ay0, py0), 0.0f);
      float inter = iw * ih;
      iou = inter / ((ax1 - ax0) * (ay1 - ay0) + areaB - inter);
    }
    if (iou > bestv) { bestv = iou; besti = t; } // first occurrence kept

    // wave32 butterfly argmax over priors for this truth (no barriers)
    float v = iou;
    int   i = valid ? p : 0x7fffffff;
    #pragma unroll
    for (int m = 16; m > 0; m >>= 1) {
      float ov = __shfl_xor(v, m, 32);
      int   oi = __shfl_xor(i, m, 32);
      if (ov > v || (ov == v && oi < i)) { v = ov; i = oi; }
    }
    if (lane == 0) { wv[t][wave] = v; wi[t][wave] = i; }
  }

  if (valid) {
    ovl[(size_t)b * P + p] = bestv;              // >= 0 (t=0 always updates)
    bti[(size_t)b * P + p] = besti;
  }

  __syncthreads();
  if (threadIdx.x < T) {                         // combine 8 wave candidates
    const int t = threadIdx.x;
    float v = wv[t][0]; int i = wi[t][0];
    #pragma unroll
    for (int w = 1; w < NWAVE; ++w) {
      float ov = wv[t][w]; int oi = wi[t][w];
      if (ov > v || (ov == v && oi < i)) { v = ov; i = oi; }
    }
    size_t idx = ((size_t)b * GX + chunk) * T + t;
    pval[idx] = v; pidx[idx] = i;
  }
}

// ---------------- kernel 2: global argmax over chunks per (b,t) ----------------
__global__ void k_bpi_reduce(const float* __restrict__ pval,
                             const int* __restrict__ pidx,
                             int* __restrict__ bpi) {
  const int bt = blockIdx.x;                     // b*T + t
  const int b = bt / T, t = bt % T;
  float v = -2.0f; int i = 0x7fffffff;
  for (int c = threadIdx.x; c < GX; c += TPB) {
    size_t idx = ((size_t)b * GX + c) * T + t;
    float ov = pval[idx]; int oi = pidx[idx];
    if (ov > v || (ov == v && oi < i)) { v = ov; i = oi; }
  }
  __shared__ float sv[TPB];
  __shared__ int   si[TPB];
  sv[threadIdx.x] = v; si[threadIdx.x] = i;
  __syncthreads();
  for (int s = TPB / 2; s > 0; s >>= 1) {
    if ((int)threadIdx.x < s) {
      float ov = sv[threadIdx.x + s]; int oi = si[threadIdx.x + s];
      if (ov > sv[threadIdx.x] ||
          (ov == sv[threadIdx.x] && oi < si[threadIdx.x])) {
        sv[threadIdx.x] = ov; si[threadIdx.x] = oi;
      }
    }
    __syncthreads();
  }
  if (threadIdx.x == 0) bpi[bt] = si[0];
}

// ---------------- kernel 3: ensure each GT keeps its best anchor ----------------
__global__ void k_override(const int* __restrict__ bpi,
                           float* __restrict__ ovl,
                           int* __restrict__ bti) {
  const int b = blockIdx.x;
  if (threadIdx.x != 0) return;
  for (int t = 0; t < T; ++t) {                  // sequential: later t wins on collision
    int idx = bpi[b * T + t];
    ovl[(size_t)b * P + idx] = 2.0f;
    bti[(size_t)b * P + idx] = t;
  }
}

// ---------------- kernel 4: fused streaming losses + mine keys ----------------
__global__ void k_main(const float* __restrict__ loc_data,
                       const float* __restrict__ conf_data,
                       const float* __restrict__ cls_data,
                       const float* __restrict__ priors,
                       const float* __restrict__ targets,
                       const float* __restrict__ ovl,
                       const int* __restrict__ bti,
                       uint32_t* __restrict__ mine,
                       int* __restrict__ num_pos,
                       float* __restrict__ part_l,
                       float* __restrict__ part_c,
                       float* __restrict__ part_id) {
  const int b = blockIdx.y;
  const int p = blockIdx.x * TPB + threadIdx.x;
  __shared__ __align__(16) float st[T * 7];
  if (threadIdx.x < (T * 7 * 4) / 16) {
    uint32_t lds = (uint32_t)(uintptr_t)(&st[0]) + threadIdx.x * 16u;
    uint64_t g   = (uint64_t)(uintptr_t)(targets + (size_t)b * T * 7) + threadIdx.x * 16u;
    async_copy_b128(lds, g);
  }
  async_wait0();
  __syncthreads();

  float ll = 0.0f, lc = 0.0f, lid = 0.0f;
  float posf = 0.0f;
  if (p < P) {
    const size_t bp = (size_t)b * P + p;
    const float o = ovl[bp];
    const int   ti = bti[bp];
    const int pos = (o < THRESH) ? 0 : 1;
    posf = (float)pos;

    const float pcx = priors[4 * p + 0], pcy = priors[4 * p + 1];
    const float pw  = priors[4 * p + 2], ph  = priors[4 * p + 3];
    const float* tr = &st[ti * 7];
    const float m0 = tr[2], m1 = tr[3], m2 = tr[4], m3 = tr[5];

    if (pos) {
      // encode(matched, prior)
      float g0 = ((m0 + m2) * 0.5f - pcx) / (VAR0 * pw);
      float g1 = ((m1 + m3) * 0.5f - pcy) / (VAR0 * ph);
      float g2 = logf((m2 - m0) / pw) / VAR1;
      float g3 = logf((m3 - m1) / ph) / VAR1;
      const float* ld = loc_data + bp * 4;
      float gt[4] = {g0, g1, g2, g3};
      #pragma unroll
      for (int j = 0; j < 4; ++j) {
        float d = ld[j] - gt[j], ad = fabsf(d);
        ll += (ad < 1.0f) ? 0.5f * d * d : ad - 0.5f;
      }
      // ID cross-entropy (ignore_index = -1)
      int cls = (int)tr[6];
      const float* xd = cls_data + bp * NID;
      float x[NID], mx = -3.0e38f;
      #pragma unroll
      for (int j = 0; j < NID; ++j) { x[j] = xd[j]; mx = fmaxf(mx, x[j]); }
      float se = 0.0f;
      #pragma unroll
      for (int j = 0; j < NID; ++j) se += expf(x[j] - mx);
      float lse = mx + logf(se);
      if (cls >= 0) {
        int sc = cls > (NID - 1) ? (NID - 1) : cls;
        lid += lse - x[sc];
      }
    }
    // confidence logits (2 classes)
    const float* cd = conf_data + bp * 2;
    float c0 = cd[0], c1 = cd[1];
    float mx2 = fmaxf(c0, c1);
    float lse2 = mx2 + logf(expf(c0 - mx2) + expf(c1 - mx2));
    if (pos) {
      lc += lse2 - c1;          // CE at class 1; positives always selected
      mine[bp] = 0u;
    } else {
      float mval = lse2 - c0;   // mining value == neg CE value (>= 0)
      mine[bp] = __float_as_uint(mval);
    }
  }

  __shared__ float red[TPB];
  float tot_l  = block_reduce_sum(ll,   red);
  float tot_c  = block_reduce_sum(lc,   red);
  float tot_id = block_reduce_sum(lid,  red);
  float tot_p  = block_reduce_sum(posf, red);   // exact small-int count in fp32
  if (threadIdx.x == 0) {
    const int idx = b * gridDim.x + blockIdx.x;
    part_l[idx]  = tot_l;
    part_c[idx]  = tot_c;
    part_id[idx] = tot_id;
    atomicAdd(&num_pos[b], (int)tot_p);         // integer atomic: deterministic
  }
}

// ---------------- kernel 5: exact top-k sum of negative mining values ----------------
// 3-level radix selection (12/12/8 bits) on the monotone uint key of mine>=0.
__global__ void k_select(const uint32_t* __restrict__ mine,
                         const int* __restrict__ num_pos,
                         float* __restrict__ negsum) {
  const int b = blockIdx.x;
  const uint32_t* key = mine + (size_t)b * P;
  __shared__ uint32_t hist[4096];
  __shared__ uint32_t s_pfx, s_rem, s_tau, s_E;
  __shared__ float red[TPB];

  long k = (long)NEG_POS * (long)num_pos[b];
  if (k > P - 1) k = P - 1;
  if (k <= 0) { if (threadIdx.x == 0) negsum[b] = 0.0f; return; }  // uniform branch

  // ---- pass 1: bits 31:20 ----
  for (int i = threadIdx.x; i < 4096; i += TPB) hist[i] = 0u;
  __syncthreads();
  for (int p = threadIdx.x; p < P; p += TPB)
    atomicAdd(&hist[key[p] >> 20], 1u);
  __syncthreads();
  if (threadIdx.x == 0) {
    uint32_t rem = (uint32_t)k; int bin = 4095;
    for (; bin > 0; --bin) { uint32_t c = hist[bin]; if (c >= rem) break; rem -= c; }
    s_pfx = (uint32_t)bin; s_rem = rem;
  }
  __syncthreads();
  const uint32_t pfx1 = s_pfx; const uint32_t rem1 = s_rem;
  __syncthreads();

  // ---- pass 2: bits 19:8 within pfx1 ----
  for (int i = threadIdx.x; i < 4096; i += TPB) hist[i] = 0u;
  __syncthreads();
  for (int p = threadIdx.x; p < P; p += TPB) {
    uint32_t kk = key[p];
    if ((kk >> 20) == pfx1) atomicAdd(&hist[(kk >> 8) & 0xFFFu], 1u);
  }
  __syncthreads();
  if (threadIdx.x == 0) {
    uint32_t rem = rem1; int bin = 4095;
    for (; bin > 0; --bin) { uint32_t c = hist[bin]; if (c >= rem) break; rem -= c; }
    s_pfx = (pfx1 << 12) | (uint32_t)bin; s_rem = rem;
  }
  __syncthreads();
  const uint32_t pfx2 = s_pfx; const uint32_t rem2 = s_rem;
  __syncthreads();

  // ---- pass 3: bits 7:0 within pfx2 ----
  for (int i = threadIdx.x; i < 256; i += TPB) hist[i] = 0u;
  __syncthreads();
  for (int p = threadIdx.x; p < P; p += TPB) {
    uint32_t kk = key[p];
    if ((kk >> 8) == pfx2) atomicAdd(&hist[kk & 0xFFu], 1u);
  }
  __syncthreads();
  if (threadIdx.x == 0) {
    uint32_t rem = rem2; int bin = 255;
    for (; bin > 0; --bin) { uint32_t c = hist[bin]; if (c >= rem) break; rem -= c; }
    s_tau = (pfx2 << 8) | (uint32_t)bin; s_E = rem;
  }
  __syncthreads();
  const uint32_t tau = s_tau; const uint32_t E = s_E;

  // ---- pass 4: sum values strictly above tau (ties contribute tau each) ----
  float acc = 0.0f;
  for (int p = threadIdx.x; p < P; p += TPB) {
    uint32_t kk = key[p];
    if (kk > tau) acc += __uint_as_float(kk);
  }
  float tot = block_reduce_sum(acc, red);
  if (threadIdx.x == 0) negsum[b] = tot + (float)E * __uint_as_float(tau);
}

// ---------------- kernel 6: deterministic finalize ----------------
__global__ void k_final(const float* __restrict__ part_l,
                        const float* __restrict__ part_c,
                        const float* __restrict__ part_id,
                        int nblk,
                        const float* __restrict__ negsum,
                        const int* __restrict__ num_pos,
                        float* __restrict__ out) {
  __shared__ float red[TPB];
  float a = 0.0f, c = 0.0f, idl = 0.0f;
  for (int j = threadIdx.x; j < nblk; j += TPB) {
    a += part_l[j]; c += part_c[j]; idl += part_id[j];
  }
  float tot_l  = block_reduce_sum(a,   red);
  float tot_c  = block_reduce_sum(c,   red);
  float tot_id = block_reduce_sum(idl, red);
  if (threadIdx.x == 0) {
    float ns = 0.0f; int npt = 0;
    for (int bb = 0; bb < B; ++bb) { ns += negsum[bb]; npt += num_pos[bb]; }
    float N = fmaxf((float)npt, 1.0f);
    out[0] = tot_l / N;
    out[1] = (tot_c + ns) / N;
    out[2] = tot_id / N;
  }
}

// ---------------- host launcher ----------------
extern "C" void kernel_launch(void* const* d_in, const int* in_sizes, int n_in,
                              void* d_out, int out_size, void* d_ws, size_t ws_size,
                              hipStream_t stream) {
  (void)in_sizes; (void)n_in; (void)out_size; (void)ws_size;
  const float* loc     = (const float*)d_in[0];
  const float* conf    = (const float*)d_in[1];
  const float* cls     = (const float*)d_in[2];
  const float* priors  = (const float*)d_in[3];
  const float* targets = (const float*)d_in[4];

  char* ws = (char*)d_ws;
  auto alignup = [](size_t x) { return (x + 255) & ~(size_t)255; };
  size_t off = 0;
  int*      bpi     = (int*)(ws + off);      off = alignup(off + sizeof(int) * B * T);
  int*      bti     = (int*)(ws + off);      off = alignup(off + sizeof(int) * (size_t)B * P);
  float*    ovl     = (float*)(ws + off);    off = alignup(off + sizeof(float) * (size_t)B * P);
  uint32_t* mine    = (uint32_t*)(ws + off); off = alignup(off + sizeof(uint32_t) * (size_t)B * P);
  int*      num_pos = (int*)(ws + off);      off = alignup(off + sizeof(int) * B);
  const int NBLK = B * GX;
  float*    part_l  = (float*)(ws + off);    off = alignup(off + sizeof(float) * NBLK);
  float*    part_c  = (float*)(ws + off);    off = alignup(off + sizeof(float) * NBLK);
  float*    part_id = (float*)(ws + off);    off = alignup(off + sizeof(float) * NBLK);
  float*    negsum  = (float*)(ws + off);    off = alignup(off + sizeof(float) * B);
  float*    pval    = (float*)(ws + off);    off = alignup(off + sizeof(float) * (size_t)B * GX * T);
  int*      pidx    = (int*)(ws + off);      off = alignup(off + sizeof(int) * (size_t)B * GX * T);

  dim3 gbp(GX, B);
  k_zero      <<<1, 64, 0, stream>>>(num_pos);
  k_match     <<<gbp, TPB, 0, stream>>>(targets, priors, ovl, bti, pval, pidx);
  k_bpi_reduce<<<B * T, TPB, 0, stream>>>(pval, pidx, bpi);
  k_override  <<<B, 32, 0, stream>>>(bpi, ovl, bti);
  k_main      <<<gbp, TPB, 0, stream>>>(loc, conf, cls, priors, targets, ovl, bti,
                                        mine, num_pos, part_l, part_c, part_id);
  k_select    <<<B, TPB, 0, stream>>>(mine, num_pos, negsum);
  k_final     <<<1, TPB, 0, stream>>>(part_l, part_c, part_id, NBLK, negsum, num_pos,
                                      (float*)d_out);
}